// PointTriangleDistance_44753559224594
// MI455X (gfx1250) — compile-verified
//
#include <hip/hip_runtime.h>
#include <math.h>

#define NT 256
#define GRID_N 32
#define NTRI 48
#define TRI_INTS (NTRI * 9)        // 432 ints
#define TRI_B128 (TRI_INTS / 4)    // 108 x 16B transfers

__device__ __forceinline__ int iclamp(int v, int lo, int hi) {
    return v < lo ? lo : (v > hi ? hi : v);
}

// reference: n / (|d| > 1e-12 ? d : 1e-12), via fast v_rcp_f32
__device__ __forceinline__ float safediv(float n, float d) {
    float dd = (__builtin_fabsf(d) > 1e-12f) ? d : 1e-12f;
    return n * __builtin_amdgcn_rcpf(dd);
}

// Ericson closest-point point-triangle squared distance, mirroring the
// reference's where-chain (later conditions override earlier ones).
__device__ __forceinline__ float tri_dist_sq(
    float px, float py, float pz,
    float ax, float ay, float az,
    float bx, float by, float bz,
    float cx, float cy, float cz)
{
    float abx = bx - ax, aby = by - ay, abz = bz - az;
    float acx = cx - ax, acy = cy - ay, acz = cz - az;
    float apx = px - ax, apy = py - ay, apz = pz - az;
    float d1 = abx*apx + aby*apy + abz*apz;
    float d2 = acx*apx + acy*apy + acz*apz;
    float bpx = px - bx, bpy = py - by, bpz = pz - bz;
    float d3 = abx*bpx + aby*bpy + abz*bpz;
    float d4 = acx*bpx + acy*bpy + acz*bpz;
    float cpx = px - cx, cpy = py - cy, cpz = pz - cz;
    float d5 = abx*cpx + aby*cpy + abz*cpz;
    float d6 = acx*cpx + acy*cpy + acz*cpz;
    float vc = d1*d4 - d3*d2;
    float vb = d5*d2 - d1*d6;
    float va = d3*d6 - d5*d4;
    float v_ab = safediv(d1, d1 - d3);
    float w_ac = safediv(d2, d2 - d6);
    float e43 = d4 - d3, e56 = d5 - d6;
    float w_bc = safediv(e43, e43 + e56);
    float inv  = safediv(1.0f, va + vb + vc);
    float v_in = vb * inv, w_in = vc * inv;
    float qx = ax + abx*v_in + acx*w_in;
    float qy = ay + aby*v_in + acy*w_in;
    float qz = az + abz*v_in + acz*w_in;
    bool m;
    m = (va <= 0.0f) & (e43 >= 0.0f) & (e56 >= 0.0f);          // edge BC
    qx = m ? bx + (cx - bx) * w_bc : qx;
    qy = m ? by + (cy - by) * w_bc : qy;
    qz = m ? bz + (cz - bz) * w_bc : qz;
    m = (vb <= 0.0f) & (d2 >= 0.0f) & (d6 <= 0.0f);            // edge AC
    qx = m ? ax + acx * w_ac : qx;
    qy = m ? ay + acy * w_ac : qy;
    qz = m ? az + acz * w_ac : qz;
    m = (d6 >= 0.0f) & (d5 <= d6);                             // vertex C
    qx = m ? cx : qx;  qy = m ? cy : qy;  qz = m ? cz : qz;
    m = (vc <= 0.0f) & (d1 >= 0.0f) & (d3 <= 0.0f);            // edge AB
    qx = m ? ax + abx * v_ab : qx;
    qy = m ? ay + aby * v_ab : qy;
    qz = m ? az + abz * v_ab : qz;
    m = (d3 >= 0.0f) & (d4 <= d3);                             // vertex B
    qx = m ? bx : qx;  qy = m ? by : qy;  qz = m ? bz : qz;
    m = (d1 <= 0.0f) & (d2 <= 0.0f);                           // vertex A
    qx = m ? ax : qx;  qy = m ? ay : qy;  qz = m ? az : qz;
    float dx = px - qx, dy = py - qy, dz = pz - qz;
    return dx*dx + dy*dy + dz*dz;
}

__global__ void __launch_bounds__(NT) zero_out_kernel(float* __restrict__ out, int n) {
    int i = blockIdx.x * NT + threadIdx.x;
    if (i < n) out[i] = 0.0f;
}

__global__ void __launch_bounds__(NT) ptd_kernel(
    const float* __restrict__ offset,   // (3,33,33,33) f32
    const float* __restrict__ points,   // (N,3) f32
    const int*   __restrict__ tri,      // (48,3,3) i32, values 0..11
    float*       __restrict__ out,      // (32768,48) f32, pre-zeroed
    int npts)
{
    // component-major vertex store: conflict-free ds_load_b32 per component
    __shared__ float vlds[36 * NT];
    __shared__ __attribute__((aligned(16))) int stri[TRI_INTS];

    const int tid = threadIdx.x;

    // ---- stage tri_table (1728 B) via CDNA5 async global->LDS DMA ----
    if (tid < TRI_B128) {
        unsigned lds_off = (unsigned)(size_t)(void*)(&stri[tid * 4]);
        const int* gp = tri + tid * 4;
        asm volatile("global_load_async_to_lds_b128 %0, %1, off"
                     :: "v"(lds_off), "v"(gp) : "memory");
    }
    asm volatile("s_wait_asynccnt 0" ::: "memory");

    const int pi = blockIdx.x * NT + tid;
    const bool valid = pi < npts;

    float plx = 0.0f, ply = 0.0f, plz = 0.0f;
    int cell = 0;
    if (valid) {
        float px = points[pi * 3 + 0];
        float py = points[pi * 3 + 1];
        float pz = points[pi * 3 + 2];
        int cx = iclamp((int)floorf(px), 0, GRID_N - 1);
        int cy = iclamp((int)floorf(py), 0, GRID_N - 1);
        int cz = iclamp((int)floorf(pz), 0, GRID_N - 1);
        plx = px - (float)cx;
        ply = py - (float)cy;
        plz = pz - (float)cz;
        cell = (cx * GRID_N + cy) * GRID_N + cz;

        // gather 12 edge offsets: offset[axis, cx+bx, cy+by, cz+bz]
        // = offset[i0 + (axis*35937 + bx*1089 + by*33 + bz)]  (constants)
        const int i0 = (cx * 33 + cy) * 33 + cz;
        constexpr int   CE[12] = {0, 37026, 33, 35937, 1, 37027, 34, 35938,
                                  71874, 72963, 72996, 71907};
        constexpr int   AXI[12] = {0, 1, 0, 1, 0, 1, 0, 1, 2, 2, 2, 2};
        constexpr float BXc[12] = {0, 1, 0, 0, 0, 1, 0, 0, 0, 1, 1, 0};
        constexpr float BYc[12] = {0, 0, 1, 0, 0, 0, 1, 0, 0, 0, 1, 1};
        constexpr float BZc[12] = {0, 0, 0, 0, 1, 1, 1, 1, 0, 0, 0, 0};
        #pragma unroll
        for (int e = 0; e < 12; ++e) {
            float t = offset[i0 + CE[e]];
            vlds[(e * 3 + 0) * NT + tid] = BXc[e] + (AXI[e] == 0 ? t : 0.0f);
            vlds[(e * 3 + 1) * NT + tid] = BYc[e] + (AXI[e] == 1 ? t : 0.0f);
            vlds[(e * 3 + 2) * NT + tid] = BZc[e] + (AXI[e] == 2 ? t : 0.0f);
        }
    }
    __syncthreads();
    if (!valid) return;

    float* orow = out + (size_t)cell * NTRI;
    for (int tt = 0; tt < NTRI; ++tt) {
        float dmin = 3.402823466e+38f;
        #pragma unroll
        for (int i = 0; i < 3; ++i) {
            // wave-uniform triangle indices -> LDS broadcast reads
            const int* tp = &stri[(tt * 3 + i) * 3];
            int ia = tp[0], ib = tp[1], ic = tp[2];
            float ax = vlds[(ia * 3 + 0) * NT + tid];
            float ay = vlds[(ia * 3 + 1) * NT + tid];
            float az = vlds[(ia * 3 + 2) * NT + tid];
            float bx = vlds[(ib * 3 + 0) * NT + tid];
            float by = vlds[(ib * 3 + 1) * NT + tid];
            float bz = vlds[(ib * 3 + 2) * NT + tid];
            float cx = vlds[(ic * 3 + 0) * NT + tid];
            float cy = vlds[(ic * 3 + 1) * NT + tid];
            float cz = vlds[(ic * 3 + 2) * NT + tid];
            float d = tri_dist_sq(plx, ply, plz, ax, ay, az, bx, by, bz, cx, cy, cz);
            dmin = fminf(dmin, d);
        }
        atomicAdd(orow + tt, dmin);   // global_atomic_add_f32, resolved in L2
    }
}

extern "C" void kernel_launch(void* const* d_in, const int* in_sizes, int n_in,
                              void* d_out, int out_size, void* d_ws, size_t ws_size,
                              hipStream_t stream) {
    (void)n_in; (void)d_ws; (void)ws_size;
    const float* offset = (const float*)d_in[0];   // (3,33,33,33)
    const float* points = (const float*)d_in[1];   // (N,3)
    const int*   tri    = (const int*)d_in[2];     // (48,3,3)
    float* out = (float*)d_out;                    // (32768,48)

    const int npts = in_sizes[1] / 3;

    zero_out_kernel<<<(out_size + NT - 1) / NT, NT, 0, stream>>>(out, out_size);
    ptd_kernel<<<(npts + NT - 1) / NT, NT, 0, stream>>>(offset, points, tri, out, npts);
}